// ReadoutInterpolator_54030688583962
// MI455X (gfx1250) — compile-verified
//
#include <hip/hip_runtime.h>

// ReadoutInterpolator: 3-tap triangular interpolation of an 8-coil complex
// k-space readout table (8 x 8192 fp32 x2) at 1e6 arbitrary positions.
// Memory-bound (~70 MB mandatory HBM traffic -> ~3us floor @ 23.3 TB/s).
// CDNA5 strategy: stage coil-pair table slices (128 KB) into LDS via the
// async global->LDS DMA path, gather from LDS, stream results out with
// non-temporal stores.

#define NOS       8192      // NRO * OVERSAMP
#define NOS_MASK  8191
#define CHUNK     4096      // points per block
#define TPB       256       // 8 waves (wave32)

typedef float v2f __attribute__((ext_vector_type(2)));
typedef float v4f __attribute__((ext_vector_type(4)));
typedef int   v4i __attribute__((ext_vector_type(4)));

#define AS_GLOBAL __attribute__((address_space(1)))
#define AS_LDS    __attribute__((address_space(3)))

typedef AS_GLOBAL v4i* g4ptr;   // builtin param 1: AS(1) int4*
typedef AS_LDS    v4i* l4ptr;   // builtin param 2: AS(3) int4* (or void*)

__device__ __forceinline__ void async_fill_b128(const float* gsrc, float* ldst) {
#if __has_builtin(__builtin_amdgcn_global_load_async_to_lds_b128)
  __builtin_amdgcn_global_load_async_to_lds_b128(
      (g4ptr)gsrc, (l4ptr)ldst, /*offset=*/0, /*cpol=*/0);
#else
  *(v4f*)ldst = *(const v4f*)gsrc;   // fallback: synchronous 128-bit copy
#endif
}

__device__ __forceinline__ void wait_async_then_barrier() {
#if __has_builtin(__builtin_amdgcn_global_load_async_to_lds_b128)
#if __has_builtin(__builtin_amdgcn_s_wait_asynccnt)
  __builtin_amdgcn_s_wait_asynccnt(0);
#else
  asm volatile("s_wait_asynccnt 0x0" ::: "memory");
#endif
#endif
  __syncthreads();
}

__global__ __launch_bounds__(TPB) void readout_interp_kernel(
    const float* __restrict__ c,
    const float* __restrict__ kspr,
    const float* __restrict__ kspi,
    v2f* __restrict__ out,
    int npts)
{
  extern __shared__ float lds[];          // 4 * NOS floats = 128 KB
  const int tid = threadIdx.x;
  const int ca  = 2 * blockIdx.y;         // this block's coil pair
  const int cb  = ca + 1;

  float* ldsRA = lds;
  float* ldsIA = lds + NOS;
  float* ldsRB = lds + 2 * NOS;
  float* ldsIB = lds + 3 * NOS;

  const float* srcs[4] = { kspr + (size_t)ca * NOS, kspi + (size_t)ca * NOS,
                           kspr + (size_t)cb * NOS, kspi + (size_t)cb * NOS };

  // Stage 4 rows (real/imag for 2 coils) into LDS: 32 async B128 copies/thread.
#pragma unroll
  for (int s = 0; s < 4; ++s) {
    const float* src = srcs[s];
    float* dst = lds + s * NOS;
    for (int j = tid * 4; j < NOS; j += TPB * 4) {
      async_fill_b128(src + j, dst + j);
    }
  }
  wait_async_then_barrier();

  const int base = blockIdx.x * CHUNK;
  for (int j = tid; j < CHUNK; j += TPB) {
    const int p = base + j;
    if (p >= npts) break;

    const float kx = c[p] * 4.0f;                 // OVERSAMP
    const float x0 = ceilf(kx - 1.0f);
    // taps x0, x0+1, x0+2 with triangular weights (match jnp exactly)
    const float w0 = fmaxf(0.0f, 1.0f - fabsf(x0 - kx));
    const float w1 = fmaxf(0.0f, 1.0f - fabsf(x0 + 1.0f - kx));
    const float w2 = fmaxf(0.0f, 1.0f - fabsf(x0 + 2.0f - kx));
    // (int)x0 truncates; x0 may be -1.0 when kx==0, & 8191 == jnp.mod(.,8192)
    const int i0 = ((int)x0) & NOS_MASK;
    const int i1 = (i0 + 1) & NOS_MASK;
    const int i2 = (i0 + 2) & NOS_MASK;

    v2f oa, ob;
    oa.x = w0 * ldsRA[i0] + w1 * ldsRA[i1] + w2 * ldsRA[i2];
    oa.y = w0 * ldsIA[i0] + w1 * ldsIA[i1] + w2 * ldsIA[i2];
    ob.x = w0 * ldsRB[i0] + w1 * ldsRB[i1] + w2 * ldsRB[i2];
    ob.y = w0 * ldsIB[i0] + w1 * ldsIB[i1] + w2 * ldsIB[i2];

    // 64 MB write-once stream: non-temporal so the ksp table stays cached.
    __builtin_nontemporal_store(oa, out + (size_t)ca * npts + p);
    __builtin_nontemporal_store(ob, out + (size_t)cb * npts + p);
  }
}

extern "C" void kernel_launch(void* const* d_in, const int* in_sizes, int n_in,
                              void* d_out, int out_size, void* d_ws, size_t ws_size,
                              hipStream_t stream) {
  const float* c    = (const float*)d_in[0];
  const float* kspr = (const float*)d_in[1];
  const float* kspi = (const float*)d_in[2];
  v2f* out = (v2f*)d_out;

  const int npts  = in_sizes[0];            // c is (NPTS,1) -> flat NPTS
  const int ncoil = in_sizes[1] / NOS;      // 8

  dim3 grid((npts + CHUNK - 1) / CHUNK, ncoil / 2, 1);
  dim3 block(TPB, 1, 1);
  const size_t shmem = 4 * NOS * sizeof(float);   // 128 KB LDS

  hipLaunchKernelGGL(readout_interp_kernel, grid, block, shmem, stream,
                     c, kspr, kspi, out, npts);
}